// LogGraphSSL_6932077215865
// MI455X (gfx1250) — compile-verified
//
#include <hip/hip_runtime.h>
#include <hip/hip_bf16.h>

typedef __attribute__((ext_vector_type(2))) float v2f;
typedef __attribute__((ext_vector_type(8))) float v8f;

// ---------------------------------------------------------------------------
// Degree / normalization kernels
// ---------------------------------------------------------------------------
__global__ void k_deg_init(float* deg, int N) {
    int i = blockIdx.x * blockDim.x + threadIdx.x;
    if (i < N) deg[i] = 1.0f;                 // self-loop contributes 1
}

__global__ void k_deg_count(const int* __restrict__ dst, float* deg, int E) {
    int e = blockIdx.x * blockDim.x + threadIdx.x;
    if (e < E) atomicAdd(&deg[dst[e]], 1.0f);
}

__global__ void k_dinv(float* deg, int N) {
    int i = blockIdx.x * blockDim.x + threadIdx.x;
    if (i < N) deg[i] = rsqrtf(deg[i]);       // deg >= 1 always
}

// ---------------------------------------------------------------------------
// Dense GEMM  Y[M,Cout] = X[M,K] @ W[K,Cout]   via V_WMMA_F32_16X16X4_F32
// One wave computes a full 16 x Cout row panel (NT = Cout/16 column tiles),
// keeping NT v8f accumulators in VGPRs. The A fragment (16x4 panel of X) is
// loaded ONCE per k-step and reused across all NT column tiles, so X is
// streamed from HBM exactly once (205 MB for layer 1 instead of 8x that).
//
// f32 16x4 A layout : lanes 0-15 -> M=lane,    VGPR{0,1} = K{0,1}
//                     lanes 16-31 -> M=lane-16, VGPR{0,1} = K{2,3}
// f32 4x16 B layout : lanes 0-15 -> N=lane,    VGPR{0,1} = K{0,1}
//                     lanes 16-31 -> N=lane-16, VGPR{0,1} = K{2,3}
// f32 16x16 C layout: VGPR v: lanes 0-15 -> M=v, lanes 16-31 -> M=v+8
// ---------------------------------------------------------------------------
template <int NT>
__global__ __launch_bounds__(32)
void k_gemm_wmma_f32(const float* __restrict__ X, const float* __restrict__ W,
                     float* __restrict__ Y, int K) {
    constexpr int Cout = NT * 16;
    const int mt   = blockIdx.x;      // row-panel index (M/16 panels)
    const int lane = threadIdx.x;     // 0..31, EXEC all ones
    const int half = lane >> 4;       // 0 or 1
    const int l16  = lane & 15;

    const float* __restrict__ xrow = X + (size_t)(mt * 16 + l16) * K;
    const float* __restrict__ wlan = W + l16;

    v8f c[NT];
#pragma unroll
    for (int t = 0; t < NT; ++t) c[t] = (v8f){};

    for (int k = 0; k < K; k += 4) {
        const int ka = k + half * 2;
        v2f a;
        a.x = xrow[ka];
        a.y = xrow[ka + 1];
        const float* __restrict__ wk = wlan + (size_t)ka * Cout;
#pragma unroll
        for (int t = 0; t < NT; ++t) {
            v2f b;
            b.x = wk[t * 16];
            b.y = wk[t * 16 + Cout];
            c[t] = __builtin_amdgcn_wmma_f32_16x16x4_f32(
                       /*neg_a=*/false, a, /*neg_b=*/false, b,
                       /*c_mod=*/(short)0, c[t],
                       /*reuse_a=*/false, /*reuse_b=*/false);
        }
    }

    float* __restrict__ yp = Y + (size_t)(mt * 16) * Cout + l16;
#pragma unroll
    for (int t = 0; t < NT; ++t)
#pragma unroll
        for (int v = 0; v < 8; ++v)
            yp[(size_t)(v + half * 8) * Cout + t * 16] = c[t][v];
}

// ---------------------------------------------------------------------------
// Aggregation: out = b + self-loop + sum over edges  (scatter via f32 atomics)
// ---------------------------------------------------------------------------

// out[i,c] = bias[c] + h[i,c] * dinv[i]^2        (grid = N blocks, block = C)
__global__ void k_agg_init(const float* __restrict__ h, const float* __restrict__ bias,
                           const float* __restrict__ dinv, float* __restrict__ out,
                           int C) {
    const int i = blockIdx.x;
    const int c = threadIdx.x;
    const float di = dinv[i];
    out[(size_t)i * C + c] = bias[c] + h[(size_t)i * C + c] * di * di;
}

// out[dst,c] += h[src,c] * dinv[src]*dinv[dst]   (grid = E blocks, block = C)
// Lanes of a wave cover consecutive channels of one edge -> coalesced gather
// and a contiguous burst of global_atomic_add_f32 that stays in L2 (h fits).
__global__ void k_agg_edges(const float* __restrict__ h,
                            const int* __restrict__ src, const int* __restrict__ dst,
                            const float* __restrict__ dinv, float* __restrict__ out,
                            int C) {
    const int e = blockIdx.x;
    const int c = threadIdx.x;
    const int s = src[e];
    const int d = dst[e];
    const float nrm = dinv[s] * dinv[d];
    atomicAdd(&out[(size_t)d * C + c], h[(size_t)s * C + c] * nrm);
}

__global__ void k_relu(float* x, long long n) {
    long long i = (long long)blockIdx.x * blockDim.x + threadIdx.x;
    if (i < n) x[i] = fmaxf(x[i], 0.0f);
}

// ---------------------------------------------------------------------------
// Orchestration
// ---------------------------------------------------------------------------
extern "C" void kernel_launch(void* const* d_in, const int* in_sizes, int n_in,
                              void* d_out, int out_size, void* d_ws, size_t ws_size,
                              hipStream_t stream) {
    const float* x  = (const float*)d_in[0];
    const int*   ei = (const int*)  d_in[1];
    const float* W1 = (const float*)d_in[2];
    const float* b1 = (const float*)d_in[3];
    const float* W2 = (const float*)d_in[4];
    const float* b2 = (const float*)d_in[5];
    const float* W3 = (const float*)d_in[6];
    const float* b3 = (const float*)d_in[7];

    const int K0 = 512, C1 = 128, C2 = 64, C3 = 32;
    const int N = in_sizes[0] / K0;           // 100000 (multiple of 16)
    const int E = in_sizes[1] / 2;            // 1.6M
    const int* src = ei;
    const int* dst = ei + E;

    // Workspace layout (floats): dinv[N] | bufH[N*128] | bufA[N*128]
    float* ws   = (float*)d_ws;
    size_t pad  = ((size_t)N + 255) & ~(size_t)255;
    float* dinv = ws;
    float* bufH = ws + pad;
    float* bufA = bufH + (size_t)N * C1;
    float* out  = (float*)d_out;

    const int TB = 256;

    // --- degrees -> dinv ---
    k_deg_init <<<(N + TB - 1) / TB, TB, 0, stream>>>(dinv, N);
    k_deg_count<<<(E + TB - 1) / TB, TB, 0, stream>>>(dst, dinv, E);
    k_dinv     <<<(N + TB - 1) / TB, TB, 0, stream>>>(dinv, N);

    // --- layer 1: h = x@W1 ; a1 = relu(agg(h) + b1) ---
    k_gemm_wmma_f32<8><<<N / 16, 32, 0, stream>>>(x, W1, bufH, K0);
    k_agg_init <<<N, C1, 0, stream>>>(bufH, b1, dinv, bufA, C1);
    k_agg_edges<<<E, C1, 0, stream>>>(bufH, src, dst, dinv, bufA, C1);
    k_relu     <<<(int)(((long long)N * C1 + TB - 1) / TB), TB, 0, stream>>>(bufA, (long long)N * C1);

    // --- layer 2: h = a1@W2 ; a2 = relu(agg(h) + b2) ---
    k_gemm_wmma_f32<4><<<N / 16, 32, 0, stream>>>(bufA, W2, bufH, C1);
    // reuse bufA for a2 (only first N*C2 floats used)
    k_agg_init <<<N, C2, 0, stream>>>(bufH, b2, dinv, bufA, C2);
    k_agg_edges<<<E, C2, 0, stream>>>(bufH, src, dst, dinv, bufA, C2);
    k_relu     <<<(int)(((long long)N * C2 + TB - 1) / TB), TB, 0, stream>>>(bufA, (long long)N * C2);

    // --- layer 3: h = a2@W3 ; out = agg(h) + b3 (no relu) ---
    k_gemm_wmma_f32<2><<<N / 16, 32, 0, stream>>>(bufA, W3, bufH, C2);
    k_agg_init <<<N, C3, 0, stream>>>(bufH, b3, dinv, out, C3);
    k_agg_edges<<<E, C3, 0, stream>>>(bufH, src, dst, dinv, out, C3);
}